// PerceiverChannelAdapter_10952166605210
// MI455X (gfx1250) — compile-verified
//
#include <hip/hip_runtime.h>

#define BATCH 16
#define CIN   128
#define TLEN  512
#define TCH   128
#define DM    64
#define EPS   1e-5f

typedef __attribute__((ext_vector_type(2))) float v2f;
typedef __attribute__((ext_vector_type(8))) float v8f;

// ---- workspace layout (float offsets) ----
#define WS_M     0                    // 64*128 folded matrix
#define WS_BVEC  8192                 // 64 folded bias
#define WS_SQ    8256                 // 128 sum_d q
#define WS_SSQ   8384                 // 128 sum_d q^2
#define WS_AQ    8512                 // 128 q . (ln_g*w_out)
#define WS_CONST 8640                 // [0]=S_w  [1]=c0
#define WS_A     8704                 // B*T*64 = 524288, a[b,t,d] row-major in d
#define WS_ST    (WS_A + 524288)      // 8192 sum_d a
#define WS_SST   (WS_ST + 8192)       // 8192 sum_d a^2
#define WS_AT    (WS_SST + 8192)      // 8192 a . (ln_g*w_out)

// ---------------------------------------------------------------------------
// Kernel 0: fold W_o*W_v*W_in -> M (64x128), fold biases -> bvec,
// per-query stats, and scalar constants. One block, tiny work.
// ---------------------------------------------------------------------------
__global__ void fold_kernel(const float* __restrict__ queries,
                            const float* __restrict__ W_in, const float* __restrict__ b_in,
                            const float* __restrict__ W_v,  const float* __restrict__ b_v,
                            const float* __restrict__ W_o,  const float* __restrict__ b_o,
                            const float* __restrict__ ln_g, const float* __restrict__ ln_b,
                            const float* __restrict__ w_out, const float* __restrict__ b_out,
                            float* __restrict__ ws) {
    __shared__ float M2s[DM * CIN];   // 32 KB of the 320 KB WGP LDS
    __shared__ float u[DM];
    const int tid = threadIdx.x;

    // M2 = W_v @ W_in  (64x128)
    for (int idx = tid; idx < DM * CIN; idx += 256) {
        const int e = idx >> 7, c = idx & 127;
        float acc = 0.f;
        for (int d = 0; d < DM; ++d) acc += W_v[e * DM + d] * W_in[d * CIN + c];
        M2s[idx] = acc;
    }
    // u[e] = W_v[e,:].b_in + b_v[e]
    if (tid < DM) {
        float acc = b_v[tid];
        for (int d = 0; d < DM; ++d) acc += W_v[tid * DM + d] * b_in[d];
        u[tid] = acc;
    }
    __syncthreads();

    // M = W_o @ M2 ;  bvec = W_o @ u + b_o
    for (int idx = tid; idx < DM * CIN; idx += 256) {
        const int f = idx >> 7, c = idx & 127;
        float acc = 0.f;
        for (int e = 0; e < DM; ++e) acc += W_o[f * DM + e] * M2s[e * CIN + c];
        ws[WS_M + idx] = acc;
    }
    if (tid < DM) {
        float acc = b_o[tid];
        for (int e = 0; e < DM; ++e) acc += W_o[tid * DM + e] * u[e];
        ws[WS_BVEC + tid] = acc;
    }
    // per-query stats
    if (tid < TCH) {
        float s = 0.f, ss = 0.f, aq = 0.f;
        for (int d = 0; d < DM; ++d) {
            const float qd = queries[tid * DM + d];
            s += qd; ss += qd * qd; aq += qd * ln_g[d] * w_out[d];
        }
        ws[WS_SQ + tid] = s; ws[WS_SSQ + tid] = ss; ws[WS_AQ + tid] = aq;
    }
    if (tid == 0) {
        float sw = 0.f, c0 = 0.f;
        for (int d = 0; d < DM; ++d) { sw += ln_g[d] * w_out[d]; c0 += ln_b[d] * w_out[d]; }
        ws[WS_CONST + 0] = sw;
        ws[WS_CONST + 1] = c0 + b_out[0];
    }
}

// ---------------------------------------------------------------------------
// Kernel 1: a[b,t,d] = sum_c M[d,c] * x[b,c,t] + bvec[d]
// One wave per 16(d) x 32(t) super-tile: two 16x16 WMMA tiles with
// independent accumulators, sharing the A (M) fragment. All loads use a
// fixed base pointer + compile-time immediate offsets.
// ---------------------------------------------------------------------------
__global__ void proj_kernel(const float* __restrict__ x, float* __restrict__ ws) {
    const float* __restrict__ M    = ws + WS_M;
    const float* __restrict__ bvec = ws + WS_BVEC;
    float* __restrict__ aout       = ws + WS_A;

    const int lane = threadIdx.x & 31;
    const int wave = threadIdx.x >> 5;
    const int tile = blockIdx.x * 8 + wave;     // 1024 super-tiles total
    const int b    = tile >> 6;                 // 16 batches
    const int r    = tile & 63;
    const int d0   = (r >> 4) * 16;             // 4 d-tiles
    const int t0   = (r & 15) * 32;             // 16 t-pairs
    const int lm   = lane & 15;
    const int half = lane >> 4;

    // lane-fixed base pointers; loop offsets become immediate offsets
    const float* __restrict__ mp = M + (d0 + lm) * CIN + 2 * half;
    const float* __restrict__ xp = x + (size_t)b * (CIN * TLEN)
                                     + (2 * half) * TLEN + t0 + lm;

    v8f acc0 = {}, acc1 = {};
    #pragma unroll 4
    for (int c0 = 0; c0 < CIN; c0 += 4) {
        const v2f av = *(const v2f*)(mp + c0);          // A: M[d0+lm][k..k+1]
        v2f bv0, bv1;
        bv0.x = xp[c0 * TLEN];                          // B: x[b][k][t]
        bv0.y = xp[c0 * TLEN + TLEN];
        bv1.x = xp[c0 * TLEN + 16];                     // B: x[b][k][t+16]
        bv1.y = xp[c0 * TLEN + TLEN + 16];
        acc0 = __builtin_amdgcn_wmma_f32_16x16x4_f32(false, av, false, bv0,
                                                     (short)0, acc0, false, false);
        acc1 = __builtin_amdgcn_wmma_f32_16x16x4_f32(false, av, false, bv1,
                                                     (short)0, acc1, false, false);
    }
    // D layout: vgpr rr, lane -> d = d0+rr+8*half, t = t0+lm (+16 for tile 1)
    const int t = t0 + lm;
    float* __restrict__ row0 = aout + (size_t)(b * TLEN + t) * DM;
    float* __restrict__ row1 = row0 + 16 * DM;
    #pragma unroll
    for (int rr = 0; rr < 8; ++rr) {
        const int d = d0 + rr + 8 * half;
        const float bb = bvec[d];
        row0[d] = acc0[rr] + bb;
        row1[d] = acc1[rr] + bb;
    }
}

// ---------------------------------------------------------------------------
// Kernel 2: per-(b,t) row stats of a: sum, sum-of-squares, a.(ln_g*w_out)
// ---------------------------------------------------------------------------
__global__ void tstat_kernel(const float* __restrict__ ln_g,
                             const float* __restrict__ w_out,
                             float* __restrict__ ws) {
    const int row = blockIdx.x * blockDim.x + threadIdx.x;   // 0..8191
    const float* __restrict__ a = ws + WS_A + (size_t)row * DM;
    float s = 0.f, ss = 0.f, aa = 0.f;
    for (int d = 0; d < DM; ++d) {
        const float v = a[d];
        s += v; ss += v * v; aa += v * ln_g[d] * w_out[d];
    }
    ws[WS_ST + row] = s; ws[WS_SST + row] = ss; ws[WS_AT + row] = aa;
}

// ---------------------------------------------------------------------------
// Kernel 3: cross GEMM C[b,t,q] = a[b,t,:].queries[q,:] via WMMA (M=q, N=t).
// One wave per 32(q) x 16(t) super-tile: two q-tiles with independent
// accumulators, sharing the B (a-row) fragment. Fused LN epilogue writes
// the (B, TC, T) output directly.
// ---------------------------------------------------------------------------
__global__ void attn_kernel(const float* __restrict__ queries,
                            const float* __restrict__ ws,
                            float* __restrict__ out) {
    const float* __restrict__ aall = ws + WS_A;

    const int lane = threadIdx.x & 31;
    const int wave = threadIdx.x >> 5;
    const int tile = blockIdx.x * 8 + wave;     // 2048 super-tiles total
    const int b    = tile >> 7;                 // 16 batches
    const int r    = tile & 127;
    const int q0   = (r >> 5) * 32;             // 4 q-pairs
    const int t0   = (r & 31) * 16;             // 32 t-tiles
    const int lm   = lane & 15;
    const int half = lane >> 4;

    const int t = t0 + lm;
    const float* __restrict__ bp  = aall + (size_t)(b * TLEN + t) * DM + 2 * half;
    const float* __restrict__ ap0 = queries + (q0 + lm) * DM + 2 * half;

    v8f acc0 = {}, acc1 = {};
    #pragma unroll 8
    for (int k0 = 0; k0 < DM; k0 += 4) {
        const v2f bv  = *(const v2f*)(bp + k0);            // shared: a[b,t,k..k+1]
        const v2f av0 = *(const v2f*)(ap0 + k0);           // queries[q0+lm]
        const v2f av1 = *(const v2f*)(ap0 + 16 * DM + k0); // queries[q0+16+lm]
        acc0 = __builtin_amdgcn_wmma_f32_16x16x4_f32(false, av0, false, bv,
                                                     (short)0, acc0, false, false);
        acc1 = __builtin_amdgcn_wmma_f32_16x16x4_f32(false, av1, false, bv,
                                                     (short)0, acc1, false, false);
    }

    const int trow = b * TLEN + t;
    const float st  = ws[WS_ST  + trow];
    const float sst = ws[WS_SST + trow];
    const float at  = ws[WS_AT  + trow];
    const float Sw  = ws[WS_CONST + 0];
    const float c0  = ws[WS_CONST + 1];
    const float inv = 1.0f / (float)DM;

    float* __restrict__ op = out + (size_t)b * (TCH * TLEN) + t;
    #pragma unroll
    for (int rr = 0; rr < 8; ++rr) {
        {   // q-tile 0: q = q0 + rr + 8*half
            const int q = q0 + rr + 8 * half;
            const float mu  = (st + ws[WS_SQ + q]) * inv;
            const float var = (sst + 2.0f * acc0[rr] + ws[WS_SSQ + q]) * inv - mu * mu;
            op[(size_t)q * TLEN] =
                rsqrtf(var + EPS) * (at + ws[WS_AQ + q] - mu * Sw) + c0;
        }
        {   // q-tile 1: q = q0 + 16 + rr + 8*half
            const int q = q0 + 16 + rr + 8 * half;
            const float mu  = (st + ws[WS_SQ + q]) * inv;
            const float var = (sst + 2.0f * acc1[rr] + ws[WS_SSQ + q]) * inv - mu * mu;
            op[(size_t)q * TLEN] =
                rsqrtf(var + EPS) * (at + ws[WS_AQ + q] - mu * Sw) + c0;
        }
    }
}

// ---------------------------------------------------------------------------
extern "C" void kernel_launch(void* const* d_in, const int* in_sizes, int n_in,
                              void* d_out, int out_size, void* d_ws, size_t ws_size,
                              hipStream_t stream) {
    const float* x       = (const float*)d_in[0];
    const float* queries = (const float*)d_in[1];
    const float* W_in    = (const float*)d_in[2];
    const float* b_in    = (const float*)d_in[3];
    const float* W_v     = (const float*)d_in[4];
    const float* b_v     = (const float*)d_in[5];
    const float* W_o     = (const float*)d_in[6];
    const float* b_o     = (const float*)d_in[7];
    const float* ln_g    = (const float*)d_in[8];
    const float* ln_b    = (const float*)d_in[9];
    const float* w_out   = (const float*)d_in[10];
    const float* b_out   = (const float*)d_in[11];
    float* ws  = (float*)d_ws;
    float* out = (float*)d_out;

    fold_kernel<<<1, 256, 0, stream>>>(queries, W_in, b_in, W_v, b_v, W_o, b_o,
                                       ln_g, ln_b, w_out, b_out, ws);
    proj_kernel<<<128, 256, 0, stream>>>(x, ws);     // 1024 waves, 2 tiles each
    tstat_kernel<<<32, 256, 0, stream>>>(ln_g, w_out, ws);
    attn_kernel<<<256, 256, 0, stream>>>(queries, ws, out);  // 2048 waves, 2 tiles each
}